// NeedlemanWunschAligner_39187281609277
// MI455X (gfx1250) — compile-verified
//
#include <hip/hip_runtime.h>
#include <hip/hip_bf16.h>

#define NEGV (-1e9f)

typedef float v2f __attribute__((ext_vector_type(2)));
typedef float v8f __attribute__((ext_vector_type(8)));

// ---------------------------------------------------------------------------
// Kernel 1: fused dual GEMM + pointwise activations, exact f32 via
// V_WMMA_F32_16X16X4_F32. One wave computes one 16x16 output tile of both
// theta (softplus) and A (log-sigmoid), looping K in steps of 4.
//
// f32 WMMA fragment layouts (ISA 7.12.2):
//   A 16x4 : lane l -> row m = l&15 ; elem e -> K = e + 2*(l>>4)
//   B 4x16 : lane l -> col n = l&15 ; elem e -> K = e + 2*(l>>4)
//   C 16x16: VGPR g -> row = g + 8*(l>>4), col = l&15
// Both A and B fragments are contiguous float2 loads from row-major zx/zy.
// ---------------------------------------------------------------------------
__global__ __launch_bounds__(256) void nw_gemm_scores(
    const float* __restrict__ zx, const float* __restrict__ zy,
    const float* __restrict__ gx, const float* __restrict__ gy,
    float* __restrict__ theta, float* __restrict__ Aout)
{
  const int lane = threadIdx.x & 31;
  const int wave = threadIdx.x >> 5;
  const int w    = blockIdx.x * 8 + wave;   // 0..8191 : b*1024 + tile
  const int b    = w >> 10;
  const int t    = w & 1023;
  const int i0   = (t >> 5) << 4;           // tile row base
  const int j0   = (t & 31) << 4;           // tile col base
  const int q    = lane & 15;               // row of A-frag == col of B-frag
  const int h    = lane >> 4;               // K half-select

  const float* za = zx + ((size_t)(b * 512 + i0 + q)) * 512 + 2 * h;
  const float* zb = zy + ((size_t)(b * 512 + j0 + q)) * 512 + 2 * h;
  const float* ga = gx + ((size_t)(b * 512 + i0 + q)) * 512 + 2 * h;
  const float* gb = gy + ((size_t)(b * 512 + j0 + q)) * 512 + 2 * h;

  v8f ct = {};  // theta accumulator
  v8f cg = {};  // gap-score accumulator
#pragma unroll 4
  for (int k = 0; k < 512; k += 4) {
    v2f a0 = *(const v2f*)(za + k);
    v2f b0 = *(const v2f*)(zb + k);
    v2f a1 = *(const v2f*)(ga + k);
    v2f b1 = *(const v2f*)(gb + k);
    ct = __builtin_amdgcn_wmma_f32_16x16x4_f32(false, a0, false, b0,
                                               (short)0, ct, false, false);
    cg = __builtin_amdgcn_wmma_f32_16x16x4_f32(false, a1, false, b1,
                                               (short)0, cg, false, false);
  }

#pragma unroll
  for (int g = 0; g < 8; ++g) {
    size_t idx = ((size_t)(b * 512 + i0 + g + 8 * h)) * 512 + j0 + q;
    float x = ct[g];
    // softplus(x) = max(x,0) + log1p(exp(-|x|))
    theta[idx] = fmaxf(x, 0.0f) + log1pf(__expf(-fabsf(x)));
    float y = cg[g];
    // log_sigmoid(y) = min(y,0) - log1p(exp(-|y|))
    Aout[idx] = fminf(y, 0.0f) - log1pf(__expf(-fabsf(y)));
  }
}

// ---------------------------------------------------------------------------
// Kernel 2: soft-NW forward + backward DP. One 512-thread workgroup per batch.
// Forward stores V (interior cells) into the aln output region; backward
// recomputes softmax weights from V/theta/A and overwrites aln with E, using
// rolling LDS diagonals so V[diag k+1,k+2] survive the in-place overwrite.
// ---------------------------------------------------------------------------
__global__ __launch_bounds__(512) void nw_dp(
    const float* __restrict__ theta, const float* __restrict__ Ag,
    float* __restrict__ aln)
{
  const int b = blockIdx.x;
  const int t = threadIdx.x;   // 0..511
  const int i = t + 1;         // DP row 1..512

  __shared__ float vb[3][513];
  __shared__ float eb[3][513];

  const float* Th = theta + (size_t)b * 512 * 512;
  const float* Aa = Ag + (size_t)b * 512 * 512;
  float* V = aln + (size_t)b * 512 * 512;   // scratch V, then final E

  // ---- forward: V[i,j] = th + lse(a+up, diag, a+left), anti-diagonal scan
  for (int x = t; x < 513; x += 512) {
    vb[0][x] = (x == 0) ? 0.0f : NEGV;  // diag 0: V[0,0]=0
    vb[1][x] = NEGV;                    // diag 1: boundary
  }
  __syncthreads();

  for (int k = 2; k <= 1024; ++k) {
    const float* p2 = vb[(k - 2) % 3];
    const float* p1 = vb[(k - 1) % 3];
    float* cur = vb[k % 3];
    int j = k - i;
    float v = NEGV;
    if (j >= 1 && j <= 512) {
      size_t o = (size_t)(i - 1) * 512 + (j - 1);
      float th = Th[o];
      float a  = Aa[o];
      float x0 = a + p1[i - 1];   // up:   A + V[i-1,j]
      float x1 = p2[i - 1];       // diag: V[i-1,j-1]
      float x2 = a + p1[i];       // left: A + V[i,j-1]
      float mx = fmaxf(fmaxf(x0, x1), x2);
      v = th + mx + __logf(__expf(x0 - mx) + __expf(x1 - mx) + __expf(x2 - mx));
      V[o] = v;
    }
    cur[i] = v;
    if (t == 0) cur[0] = NEGV;    // V[0,j] boundary
    __syncthreads();
  }

  __threadfence();
  __syncthreads();

  // ---- backward: E[i,j] = sum over children of weight * E[child]
  // weights recomputed: w = exp(A + theta(child) + V[i,j] - V[child])
  for (int x = t; x < 513; x += 512) {
    vb[2][x] = 0.0f; eb[2][x] = 0.0f;   // diag 1025 (1025%3==2)
    vb[0][x] = 0.0f; eb[0][x] = 0.0f;   // diag 1026 (1026%3==0)
  }
  __syncthreads();

  for (int k = 1024; k >= 2; --k) {
    const float* v1 = vb[(k + 1) % 3];
    const float* v2 = vb[(k + 2) % 3];
    float* vcur = vb[k % 3];
    const float* e1 = eb[(k + 1) % 3];
    const float* e2 = eb[(k + 2) % 3];
    float* ecur = eb[k % 3];
    int j = k - i;
    float e = 0.0f, vij = 0.0f;
    if (j >= 1 && j <= 512) {
      size_t o = (size_t)(i - 1) * 512 + (j - 1);
      vij = V[o];
      if (i < 512) {            // child (i+1, j), "up" edge
        size_t o1 = (size_t)i * 512 + (j - 1);
        e += __expf(Aa[o1] + Th[o1] + vij - v1[i + 1]) * e1[i + 1];
      }
      if (i < 512 && j < 512) { // child (i+1, j+1), "diag" edge
        size_t o2 = (size_t)i * 512 + j;
        e += __expf(Th[o2] + vij - v2[i + 1]) * e2[i + 1];
      }
      if (j < 512) {            // child (i, j+1), "left" edge
        size_t o3 = (size_t)(i - 1) * 512 + j;
        e += __expf(Aa[o3] + Th[o3] + vij - v1[i]) * e1[i];
      }
      if (i == 512 && j == 512) e = 1.0f;  // terminal seed
      V[o] = e;                 // overwrite V with alignment output
    }
    vcur[i] = vij;
    ecur[i] = e;
    if (t == 0) { vcur[0] = 0.0f; ecur[0] = 0.0f; }
    __syncthreads();
  }
}

extern "C" void kernel_launch(void* const* d_in, const int* in_sizes, int n_in,
                              void* d_out, int out_size, void* d_ws, size_t ws_size,
                              hipStream_t stream) {
  (void)in_sizes; (void)n_in; (void)out_size; (void)d_ws; (void)ws_size;
  const float* zx = (const float*)d_in[0];
  const float* zy = (const float*)d_in[1];
  const float* gx = (const float*)d_in[2];
  const float* gy = (const float*)d_in[3];

  float* out   = (float*)d_out;
  const size_t plane = (size_t)8 * 512 * 512;
  float* aln   = out;               // output 0
  float* theta = out + plane;       // output 1
  float* Amat  = out + 2 * plane;   // output 2

  // 8192 tiles (16x16) of theta & A: 8 waves per 256-thread block
  nw_gemm_scores<<<1024, 256, 0, stream>>>(zx, zy, gx, gy, theta, Amat);
  // one workgroup per batch element for the sequential wavefront DP
  nw_dp<<<8, 512, 0, stream>>>(theta, Amat, aln);
}